// SpmvCudaWrapper_78597901516910
// MI455X (gfx1250) — compile-verified
//
#include <hip/hip_runtime.h>
#include <stdint.h>

// CSR SpMV for MI455X (gfx1250), nonzero-split scheme.
// Memory-bound: ~264 MB of traffic -> ~11 us at 23.3 TB/s.
//   - sx streamed once via CDNA5 async global->LDS B128 DMA (NT hint),
//     overlapped with the sel (B128, NT) load + x gather (x = 4 MB, kept
//     L2-resident because both big streams are marked non-temporal).
//   - products land in LDS (B128 RMW); per-row segmented reduction;
//     <=2 commutative f32 atomics per row (max row length << CHUNK)
//     -> bitwise-deterministic output.

#define BLOCK 256
#define CHUNK 4096              // nonzeros per block; 16 KB LDS
#define VEPT  (CHUNK / (BLOCK * 4))  // float4-granule iterations per thread (= 4)

typedef __attribute__((ext_vector_type(4))) int   v4i;
typedef __attribute__((ext_vector_type(4))) float v4f;

__global__ __launch_bounds__(BLOCK) void spmv_zero_y(float* __restrict__ y, int n) {
  int i = blockIdx.x * blockDim.x + threadIdx.x;
  if (i < n) y[i] = 0.0f;
}

__global__ __launch_bounds__(BLOCK) void spmv_chunk_kernel(
    const float* __restrict__ sx, const float* __restrict__ x,
    const int* __restrict__ sel, const int* __restrict__ ro,
    float* __restrict__ y, int nnz, int num_rows) {
  __shared__ __align__(16) float s_val[CHUNK];
  __shared__ int s_row_range[2];

  const int c0  = blockIdx.x * CHUNK;
  const int len = min(CHUNK, nnz - c0);
  const int tid = threadIdx.x;

  // ---- Phase 1: kick off async DMA of the sx chunk into LDS.
  // Each b128 issue moves 16B/lane (512 B/wave-instr), fully coalesced,
  // non-temporal (sx is streamed exactly once). Tracked by ASYNCcnt.
  {
    const float* gsx = sx + c0;
#pragma unroll
    for (int k = 0; k < VEPT; ++k) {
      int e = (k * BLOCK + tid) * 4;
      if (e + 4 <= len) {
        // Low 32 bits of a flat LDS pointer are the LDS byte address (ISA 10.2).
        uint32_t lv = (uint32_t)(uintptr_t)(&s_val[e]);
        uint64_t gv = (uint64_t)(uintptr_t)(gsx + e);
        asm volatile("global_load_async_to_lds_b128 %0, %1, off th:TH_LOAD_NT"
                     :: "v"(lv), "v"(gv) : "memory");
      } else {
        // Ragged tail (last chunk only): guarded scalar copy.
        for (int t = 0; t < 4; ++t) {
          int e2 = e + t;
          if (e2 < len) s_val[e2] = gsx[e2];
        }
      }
    }
  }

  // ---- Overlapped with the DMA: rows intersecting [c0, c0+len).
  // rlo = row containing c0, rhi = row containing c0+len-1
  // (searchsorted(ro, t, 'right') - 1; matches reference row_ids semantics).
  if (tid < 2) {
    int target = (tid == 0) ? c0 : (c0 + len - 1);
    int lo = 0, hi = num_rows;  // ro[0]=0 <= target < ro[num_rows]=nnz
    while (hi - lo > 1) {
      int mid = (lo + hi) >> 1;
      if (ro[mid] <= target) lo = mid; else hi = mid;
    }
    s_row_range[tid] = lo;
  }

  // ---- Phase 2a: also overlapped with the DMA: load sel in B128 granules
  // (NT, streamed once) and gather x into registers. x is 4 MB -> L2-resident.
  v4f xv[VEPT];
#pragma unroll
  for (int k = 0; k < VEPT; ++k) {
    int e = (k * BLOCK + tid) * 4;
    xv[k] = (v4f)0.0f;
    if (e + 4 <= len) {
      v4i c = __builtin_nontemporal_load((const v4i*)(sel + c0 + e));
      xv[k].x = x[c.x];
      xv[k].y = x[c.y];
      xv[k].z = x[c.z];
      xv[k].w = x[c.w];
    } else {
      for (int t = 0; t < 4; ++t) {
        int e2 = e + t;
        if (e2 < len) xv[k][t] = x[sel[c0 + e2]];
      }
    }
  }

  // DMA completion, then cross-wave visibility of s_val.
  asm volatile("s_wait_asynccnt 0x0" ::: "memory");
  __syncthreads();

  // ---- Phase 2b: products in place in LDS, B128 granules.
  {
    v4f* sv4 = (v4f*)s_val;
#pragma unroll
    for (int k = 0; k < VEPT; ++k) {
      int e = (k * BLOCK + tid) * 4;
      if (e + 4 <= len) {
        v4f v = sv4[k * BLOCK + tid];
        v *= xv[k];
        sv4[k * BLOCK + tid] = v;
      } else {
        for (int t = 0; t < 4; ++t) {
          int e2 = e + t;
          if (e2 < len) s_val[e2] *= xv[k][t];
        }
      }
    }
  }
  __syncthreads();

  // ---- Phase 3: per-row segmented reduction, one thread per row in range.
  // Max row length << CHUNK => each row spans <=2 chunks => <=2 commutative
  // f32 atomic adds => bitwise-deterministic output.
  const int rlo = s_row_range[0];
  const int rhi = s_row_range[1];
  for (int r = rlo + tid; r <= rhi; r += BLOCK) {
    int s = ro[r], e = ro[r + 1];
    s = max(s, c0);
    e = min(e, c0 + len);
    if (s < e) {
      float acc = 0.0f;
      for (int i = s - c0; i < e - c0; ++i) acc += s_val[i];
      atomicAdd(&y[r], acc);  // global_atomic_add_f32 (no return)
    }
  }
}

extern "C" void kernel_launch(void* const* d_in, const int* in_sizes, int n_in,
                              void* d_out, int out_size, void* d_ws, size_t ws_size,
                              hipStream_t stream) {
  const float* sx  = (const float*)d_in[0];
  const float* x   = (const float*)d_in[1];
  // d_in[2] = y input (unused by the reference computation)
  const int*   sel = (const int*)d_in[3];
  const int*   ro  = (const int*)d_in[4];
  float* y = (float*)d_out;

  const int nnz      = in_sizes[0];
  const int num_rows = in_sizes[4] - 1;

  spmv_zero_y<<<(num_rows + BLOCK - 1) / BLOCK, BLOCK, 0, stream>>>(y, num_rows);

  const int nblocks = (nnz + CHUNK - 1) / CHUNK;
  spmv_chunk_kernel<<<nblocks, BLOCK, 0, stream>>>(sx, x, sel, ro, y, nnz, num_rows);
}